// DecompGridv3_78099685310705
// MI455X (gfx1250) — compile-verified
//
#include <hip/hip_runtime.h>
#include <hip/hip_bf16.h>

// ---------------------------------------------------------------------------
// DecompGridv3 sampler for MI455X (gfx1250, wave32).
//
// Phase 1: transpose all factor grids from feature-outer (F, N) to
//          feature-inner (N, F=32) layout in d_ws, using
//          V_WMMA_F32_16X16X4_F32 with shifted identity B-matrices as an
//          in-register 16x16 transpose engine (A-layout in, C-layout out;
//          the register-layout difference performs the transpose, no LDS).
// Phase 2: gather/interp kernel, lane==feature, so every 32-feature corner
//          fetch is ONE fully-coalesced 128B cache line.
//
// d_ws requirement: 32*(128^3 + 3*384^2 + 256) floats ~= 325 MB.
// ---------------------------------------------------------------------------

typedef float v2f __attribute__((ext_vector_type(2)));
typedef float v8f __attribute__((ext_vector_type(8)));

#define NFEAT 32
#define D3    128
#define P2    384
#define L1    256
#define NPTS  (1 << 20)

// Transpose (32, N) row-major -> (N, 32) row-major via WMMA identity-multiply.
// Each wave handles one 16-wide n-tile across all 32 features per iteration.
//
// A_c[m][k] = in[f0 + 4c + k][n0 + m]   (A layout: lane(0..15)->M, VGPRs->K,
//                                        lanes 16..31 hold K+2 halves)
// B_c      = shifted identity, B_c[k][n] = (n == 4c + k)
// D[m][n]  = sum_c A_c x B_c = in[f0+n][n0+m]  -> out[(n0+m)*32 + f0+n]
__global__ void wmma_transpose_32xN(const float* __restrict__ in,
                                    float* __restrict__ out,
                                    long N) {
  const int lane  = threadIdx.x & 31;
  const int wave  = threadIdx.x >> 5;
  const int wpb   = blockDim.x >> 5;
  const long numTiles  = N >> 4;                       // N is a multiple of 16
  const long waveId    = (long)blockIdx.x * wpb + wave;
  const long waveCount = (long)gridDim.x * wpb;

  const int m     = lane & 15;
  const int khalf = (lane >> 4) << 1;                  // 0 or 2

  for (long t = waveId; t < numTiles; t += waveCount) {
    const long n0 = t << 4;
#pragma unroll
    for (int f0 = 0; f0 < NFEAT; f0 += 16) {
      v8f acc = {0.f, 0.f, 0.f, 0.f, 0.f, 0.f, 0.f, 0.f};
#pragma unroll
      for (int c = 0; c < 4; ++c) {
        v2f a;
        const float* src = in + (long)(f0 + 4 * c + khalf) * N + n0 + m;
        a.x = src[0];                                  // K = khalf
        a.y = src[N];                                  // K = khalf + 1
        v2f b;                                         // shifted identity
        b.x = (m == (4 * c + khalf))     ? 1.0f : 0.0f;
        b.y = (m == (4 * c + khalf + 1)) ? 1.0f : 0.0f;
        acc = __builtin_amdgcn_wmma_f32_16x16x4_f32(
            false, a, false, b, (short)0, acc, false, false);
      }
      // C/D layout: VGPR g -> rows {g, g+8}, N across lanes.
#pragma unroll
      for (int g = 0; g < 8; ++g) {
        const long row = n0 + g + ((lane >> 4) << 3);
        out[row * NFEAT + f0 + m] = acc[g];
      }
    }
  }
}

__device__ __forceinline__ void corner1d(float f, int R, int& i0, int& i1, float& w) {
  float f0 = floorf(f);
  w = f - f0;
  float c = fminf(fmaxf(f0, 0.0f), (float)(R - 1));
  i0 = (int)c;
  i1 = min(i0 + 1, R - 1);
}

__global__ void __launch_bounds__(256)
decomp_sample_kernel(const float4* __restrict__ x,
                     const float* __restrict__ gT,
                     const float* __restrict__ p01T,
                     const float* __restrict__ p02T,
                     const float* __restrict__ p12T,
                     const float* __restrict__ lineT,
                     float* __restrict__ out) {
  const int lane = threadIdx.x & 31;           // lane == feature index
  const int wave = threadIdx.x >> 5;
  const long waveId    = (long)blockIdx.x * (blockDim.x >> 5) + wave;
  const long waveCount = (long)gridDim.x * (blockDim.x >> 5);

  for (long b = waveId; b < NPTS; b += waveCount) {
    // Prefetch next point's coordinate row (global_prefetch_b8).
    __builtin_prefetch((const void*)(x + b + waveCount), 0, 0);

    const float4 p = x[b];                     // uniform across the wave

    // ---- 3D trilinear (grid 128^3, cz->D, cy->H, cx->W) ----
    int x0, x1, y0, y1, z0, z1; float wx, wy, wz;
    corner1d((p.x + 1.0f) * 0.5f * (float)(D3 - 1), D3, x0, x1, wx);
    corner1d((p.y + 1.0f) * 0.5f * (float)(D3 - 1), D3, y0, y1, wy);
    corner1d((p.z + 1.0f) * 0.5f * (float)(D3 - 1), D3, z0, z1, wz);

    #define G3(zi, yi, xi) gT[((((long)(zi) * D3 + (yi)) * D3 + (xi)) << 5) + lane]
    float g000 = G3(z0, y0, x0), g001 = G3(z0, y0, x1);
    float g010 = G3(z0, y1, x0), g011 = G3(z0, y1, x1);
    float g100 = G3(z1, y0, x0), g101 = G3(z1, y0, x1);
    float g110 = G3(z1, y1, x0), g111 = G3(z1, y1, x1);
    #undef G3
    float c00 = g000 + wx * (g001 - g000);
    float c01 = g010 + wx * (g011 - g010);
    float c10 = g100 + wx * (g101 - g100);
    float c11 = g110 + wx * (g111 - g110);
    float c0  = c00 + wy * (c01 - c00);
    float c1  = c10 + wy * (c11 - c10);
    float f3  = c0 + wz * (c1 - c0);

    // ---- 2D bilinear planes, 384^2 (cx->W, cy->H) ----
    #define G2(P, yi, xi) (P)[((((long)(yi) * P2) + (xi)) << 5) + lane]
    int a0, a1, b0, b1; float wa, wb;
    // plane01: (cx = p.x, cy = p.y)
    corner1d((p.x + 1.0f) * 0.5f * (float)(P2 - 1), P2, a0, a1, wa);
    corner1d((p.y + 1.0f) * 0.5f * (float)(P2 - 1), P2, b0, b1, wb);
    float q00 = G2(p01T, b0, a0), q01v = G2(p01T, b0, a1);
    float q10 = G2(p01T, b1, a0), q11 = G2(p01T, b1, a1);
    float r0 = q00 + wa * (q01v - q00);
    float r1 = q10 + wa * (q11 - q10);
    float f01 = r0 + wb * (r1 - r0);
    // plane02: (cx = p.x, cy = p.z)
    corner1d((p.z + 1.0f) * 0.5f * (float)(P2 - 1), P2, b0, b1, wb);
    q00 = G2(p02T, b0, a0); q01v = G2(p02T, b0, a1);
    q10 = G2(p02T, b1, a0); q11 = G2(p02T, b1, a1);
    r0 = q00 + wa * (q01v - q00);
    r1 = q10 + wa * (q11 - q10);
    float f02 = r0 + wb * (r1 - r0);
    // plane12: (cx = p.y, cy = p.z)
    corner1d((p.y + 1.0f) * 0.5f * (float)(P2 - 1), P2, a0, a1, wa);
    q00 = G2(p12T, b0, a0); q01v = G2(p12T, b0, a1);
    q10 = G2(p12T, b1, a0); q11 = G2(p12T, b1, a1);
    r0 = q00 + wa * (q01v - q00);
    r1 = q10 + wa * (q11 - q10);
    float f12 = r0 + wb * (r1 - r0);
    #undef G2

    // ---- 1D line lerp (L1 = 256) ----
    float xn  = p.w * (float)L1;
    float i0f = floorf(xn);
    float wl  = xn - i0f;
    int i0 = (int)i0f;
    int i1 = (int)fminf(fmaxf(i0f + 1.0f, 0.0f), (float)(L1 - 1));
    float l0 = lineT[((long)i0 << 5) + lane];
    float l1v = lineT[((long)i1 << 5) + lane];
    float fl = l0 + wl * (l1v - l0);

    out[(b << 5) + lane] = f3 * f01 * f02 * f12 * fl;
  }
}

extern "C" void kernel_launch(void* const* d_in, const int* in_sizes, int n_in,
                              void* d_out, int out_size, void* d_ws, size_t ws_size,
                              hipStream_t stream) {
  const float* x       = (const float*)d_in[0];  // (B, 4)
  const float* grid3d  = (const float*)d_in[1];  // (32, 128,128,128)
  const float* plane01 = (const float*)d_in[2];  // (32, 384,384)
  const float* plane02 = (const float*)d_in[3];
  const float* plane12 = (const float*)d_in[4];
  const float* line0   = (const float*)d_in[5];  // (32, 256)
  float* out = (float*)d_out;

  const long N3 = (long)D3 * D3 * D3;   // 2,097,152
  const long N2 = (long)P2 * P2;        // 147,456
  const long N1 = L1;                   // 256

  // Workspace partition (feature-inner copies), ~325 MB total.
  float* ws   = (float*)d_ws;
  float* gT   = ws;                       ws += (size_t)NFEAT * N3;
  float* p01T = ws;                       ws += (size_t)NFEAT * N2;
  float* p02T = ws;                       ws += (size_t)NFEAT * N2;
  float* p12T = ws;                       ws += (size_t)NFEAT * N2;
  float* lnT  = ws;

  // Phase 1: WMMA-based feature transposes.
  wmma_transpose_32xN<<<4096, 256, 0, stream>>>(grid3d,  gT,   N3);
  wmma_transpose_32xN<<<1024, 256, 0, stream>>>(plane01, p01T, N2);
  wmma_transpose_32xN<<<1024, 256, 0, stream>>>(plane02, p02T, N2);
  wmma_transpose_32xN<<<1024, 256, 0, stream>>>(plane12, p12T, N2);
  wmma_transpose_32xN<<<2,    256, 0, stream>>>(line0,   lnT,  N1);

  // Phase 2: coalesced gather + interpolate + product.
  decomp_sample_kernel<<<4096, 256, 0, stream>>>(
      (const float4*)x, gT, p01T, p02T, p12T, lnT, out);
}